// CropAndResize_60146722013533
// MI455X (gfx1250) — compile-verified
//
#include <hip/hip_runtime.h>

typedef float v2f __attribute__((ext_vector_type(2)));
typedef float v8f __attribute__((ext_vector_type(8)));

#define CROP_H  14
#define CROP_W  14
#define IMG_H   200
#define IMG_W   200
#define NCH     256
#define NCOL    28              // 2 tap columns per crop x
#define KSTEPS  7               // 28 / 4
#define VROW    (NCH*4 + 8)     // 1032 floats per k-group; +8 pad => conflict-free banks
#define VBUF    (KSTEPS * VROW) // 7224 floats per buffer

static __device__ __forceinline__ v2f mk2(float a, float b) { v2f r; r.x = a; r.y = b; return r; }

// out(14x256 per crop row) = W(14x28, 2 nonzeros/row) * V(28x256)
// as 7 k-steps of V_WMMA_F32_16X16X4_F32 per 16x16 tile.
__global__ __launch_bounds__(256)
void crop_resize_wmma(const float* __restrict__ image,
                      const float* __restrict__ boxes,
                      const int*   __restrict__ box_idx,
                      float* __restrict__ out)
{
    __shared__ __align__(16) float sV[2][VBUF];        // V[kk][ch][4] double-buffered
    __shared__ float swy0[CROP_H], swy1[CROP_H];
    __shared__ int   sy0[CROP_H],  sy1[CROP_H];
    __shared__ float swx0[CROP_W], swx1[CROP_W];
    __shared__ int   sx[NCOL];                         // sx[2i]=x0_i, sx[2i+1]=x1_i

    const int n    = blockIdx.x;
    const int tid  = threadIdx.x;
    const int wave = tid >> 5;
    const int lane = tid & 31;

    // ---- per-box sampling coordinates (threads 0..13) ----
    if (tid < CROP_H) {
        const float by1 = boxes[n*4+0], bx1 = boxes[n*4+1];
        const float by2 = boxes[n*4+2], bx2 = boxes[n*4+3];

        const float hs = (by2 - by1) * (float)(IMG_H-1) / (float)(CROP_H-1);
        const float iy = by1 * (float)(IMG_H-1) + (float)tid * hs;
        const bool  vy = (iy >= 0.f) && (iy <= (float)(IMG_H-1));
        const float fy = floorf(iy);
        const float ly = iy - fy;
        sy0[tid]  = min(max((int)fy, 0), IMG_H-1);
        sy1[tid]  = min(max((int)ceilf(iy), 0), IMG_H-1);
        swy0[tid] = vy ? (1.f - ly) : 0.f;
        swy1[tid] = vy ? ly : 0.f;

        const float ws = (bx2 - bx1) * (float)(IMG_W-1) / (float)(CROP_W-1);
        const float ix = bx1 * (float)(IMG_W-1) + (float)tid * ws;
        const bool  vx = (ix >= 0.f) && (ix <= (float)(IMG_W-1));
        const float fx = floorf(ix);
        const float lx = ix - fx;
        sx[2*tid+0] = min(max((int)fx, 0), IMG_W-1);
        sx[2*tid+1] = min(max((int)ceilf(ix), 0), IMG_W-1);
        swx0[tid]   = vx ? (1.f - lx) : 0.f;
        swx1[tid]   = vx ? lx : 0.f;
    }
    __syncthreads();

    const float* imgb = image + (size_t)box_idx[n] * (size_t)(NCH * IMG_H * IMG_W);
    const int half = lane >> 4;        // WMMA half-wave (K split / D row split)
    const int l16  = lane & 15;
    const int m    = l16;              // A-matrix row = output crop_x

    // ---- A fragments: invariant over crop rows and channel tiles -> registers ----
    const float wxa = (m < CROP_W) ? swx0[m] : 0.f;
    const float wxb = (m < CROP_W) ? swx1[m] : 0.f;
    v2f afrag[KSTEPS];
    #pragma unroll
    for (int kk = 0; kk < KSTEPS; ++kk) {
        const int k0 = kk*4 + half*2;
        const int k1 = k0 + 1;
        afrag[kk].x = (k0 == 2*m) ? wxa : ((k0 == 2*m+1) ? wxb : 0.f);
        afrag[kk].y = (k1 == 2*m) ? wxa : ((k1 == 2*m+1) ? wxb : 0.f);
    }

    const int g = lane >> 2, r = lane & 3;   // V[kk=g][ch][r] slot for tap column `lane`

    for (int cy = 0; cy < CROP_H; ++cy) {
        float* V = sV[cy & 1];

        // ---- Stage A: vertical lerp gather -> LDS ----
        // lanes index the 28 tap columns (clustered inside one image row: few lines/wave),
        // waves stride over channels; prefetch next crop row's tap rows.
        if (lane < NCOL) {
            const int   xcol = sx[lane];
            const int   o0 = sy0[cy] * IMG_W + xcol;
            const int   o1 = sy1[cy] * IMG_W + xcol;
            const int   cyn = (cy + 1 < CROP_H) ? cy + 1 : cy;
            const int   f0 = sy0[cyn] * IMG_W + xcol;
            const int   f1 = sy1[cyn] * IMG_W + xcol;
            const float w0 = swy0[cy], w1 = swy1[cy];
            float* Vg = V + g * VROW + r;
            for (int c = wave; c < NCH; c += 8) {
                const float* p = imgb + (size_t)c * (IMG_H * IMG_W);
                Vg[c * 4] = w0 * p[o0] + w1 * p[o1];
                __builtin_prefetch(p + f0, 0, 3);   // global_prefetch_b8: next row taps
                __builtin_prefetch(p + f1, 0, 3);
            }
        }
        __syncthreads();   // single barrier per iteration (V double-buffered)

        // ---- Stage B: 7 WMMAs per 16x16 tile, B via single ds_load_b64 per step ----
        const v2f* Vb = (const v2f*)V;
        for (int t = 0; t < 2; ++t) {              // 8 waves x 2 = 16 channel tiles
            const int chbase = (wave + t*8) * 16;
            const int boff   = ((chbase + l16) * 4 + half * 2) >> 1;   // v2f units
            v8f acc = {};
            #pragma unroll
            for (int kk = 0; kk < KSTEPS; ++kk) {
                const v2f b = Vb[kk * (VROW >> 1) + boff];
                acc = __builtin_amdgcn_wmma_f32_16x16x4_f32(
                          false, afrag[kk], false, b, (short)0, acc, false, false);
            }
            // D rows = contiguous crop_x; row base is 8B-aligned (196 even) -> b64 stores
            const int ch = chbase + l16;
            v2f* o = (v2f*)(out + (((size_t)n * NCH + ch) * CROP_H + cy) * CROP_W);
            if (half == 0) {
                o[0] = mk2(acc[0], acc[1]);
                o[1] = mk2(acc[2], acc[3]);
                o[2] = mk2(acc[4], acc[5]);
                o[3] = mk2(acc[6], acc[7]);
            } else {                               // m = 8..13 valid
                o[4] = mk2(acc[0], acc[1]);
                o[5] = mk2(acc[2], acc[3]);
                o[6] = mk2(acc[4], acc[5]);
            }
        }
    }
}

extern "C" void kernel_launch(void* const* d_in, const int* in_sizes, int n_in,
                              void* d_out, int out_size, void* d_ws, size_t ws_size,
                              hipStream_t stream) {
    const float* image = (const float*)d_in[0];
    const float* boxes = (const float*)d_in[1];
    const int*   bidx  = (const int*)d_in[2];
    float*       outp  = (float*)d_out;
    const int nboxes = in_sizes[2];   // 512
    crop_resize_wmma<<<nboxes, 256, 0, stream>>>(image, boxes, bidx, outp);
}